// GraphConvolution_81990925681141
// MI455X (gfx1250) — compile-verified
//
#include <hip/hip_runtime.h>

#define D_FEAT 128
#define NEG_SLOPE 0.01f

typedef __attribute__((ext_vector_type(2))) float v2f;
typedef __attribute__((ext_vector_type(8))) float v8f;

// ---------------------------------------------------------------------------
// Kernel 1: out[node][j] = bias[j]   (folds the bias add into the init,
// so the scatter-adds accumulate on top of it)
// ---------------------------------------------------------------------------
__global__ void __launch_bounds__(256) init_bias_kernel(float* __restrict__ out,
                                                        const float* __restrict__ bias,
                                                        int n4 /* n_nodes * 32 */) {
    int i = blockIdx.x * blockDim.x + threadIdx.x;
    if (i >= n4) return;
    float4 b = ((const float4*)bias)[i & 31];   // 128 floats = 32 float4, broadcast
    ((float4*)out)[i] = b;
}

// ---------------------------------------------------------------------------
// Kernel 2: H = X @ W  via V_WMMA_F32_16X16X4_F32 (fp32 exact, K=128 = 32 steps)
// Block = 256 threads = 8 waves. Block computes a 16-row x 128-col slab of H.
// A tile (16x128) staged in LDS with stride 132 (bank-conflict padding);
// each wave computes one 16x16 tile (its own 16-col slice of W).
// ---------------------------------------------------------------------------
__global__ void __launch_bounds__(256) gemm_wmma_kernel(const float* __restrict__ X,
                                                        const float* __restrict__ W,
                                                        float* __restrict__ H) {
    __shared__ float As[16 * 132];              // 16 rows, padded stride 132 (8.25 KB)
    const int t    = threadIdx.x;
    const int row0 = blockIdx.x * 16;

    // Cooperative stage of the 16x128 A tile: 512 float4 chunks, 2 per thread.
    {
        const float4* Xv = (const float4*)(X + (size_t)row0 * D_FEAT);
        #pragma unroll
        for (int i = 0; i < 2; ++i) {
            int idx = t + i * 256;              // 0..511
            int r   = idx >> 5;                 // tile row 0..15
            int c4  = idx & 31;                 // float4 column
            float4 v = Xv[idx];
            ((float4*)(As + r * 132))[c4] = v;  // 528B row stride: 16B aligned
        }
    }
    __syncthreads();

    const int wave = t >> 5;
    const int lane = t & 31;
    const int m    = lane & 15;                 // matrix row (A) / col (B,D)
    const int hi   = lane >> 4;                 // half-wave select
    const int n    = wave * 16 + m;             // output column owned by this lane

    v8f acc = {};
    #pragma unroll
    for (int k4 = 0; k4 < 32; ++k4) {
        const int kk = k4 * 4 + hi * 2;         // lanes 0-15: K=0,1  lanes 16-31: K=2,3
        // A fragment: row m, K = kk, kk+1 (8B-aligned LDS float2 load)
        v2f a = *(const v2f*)(As + m * 132 + kk);
        // B fragment: W[kk][n], W[kk+1][n] — 16-lane contiguous 64B per row, L2-hot
        v2f b;
        b.x = W[kk * D_FEAT + n];
        b.y = W[(kk + 1) * D_FEAT + n];
        acc = __builtin_amdgcn_wmma_f32_16x16x4_f32(false, a, false, b,
                                                    (short)0, acc, false, false);
    }

    // D layout: VGPR r -> M = r (lanes 0-15) / M = 8+r (lanes 16-31), N = n
    #pragma unroll
    for (int r = 0; r < 8; ++r) {
        int mrow = hi * 8 + r;
        H[(size_t)(row0 + mrow) * D_FEAT + n] = acc[r];
    }
}

// ---------------------------------------------------------------------------
// Kernel 3: edge-parallel SpMM scatter. One wave per edge (grid-stride):
// 32 lanes x float4 = one coalesced 512B row gather from H (L2-resident),
// scaled by edge_val, then hardware f32 atomic adds into out (L2 atomics).
// ---------------------------------------------------------------------------
__global__ void __launch_bounds__(256) spmm_scatter_kernel(const float* __restrict__ H,
                                                           const int* __restrict__ erow,
                                                           const int* __restrict__ ecol,
                                                           const float* __restrict__ eval_,
                                                           float* out, int n_edges) {
    const int lane   = threadIdx.x & 31;
    const int wave   = blockIdx.x * (blockDim.x >> 5) + (threadIdx.x >> 5);
    const int nwaves = gridDim.x * (blockDim.x >> 5);

    for (int e = wave; e < n_edges; e += nwaves) {
        const int   r = erow[e];                // uniform across the wave
        const int   c = ecol[e];
        const float v = eval_[e];

        float4 m4 = ((const float4*)(H + (size_t)c * D_FEAT))[lane];
        float* o  = out + (size_t)r * D_FEAT + lane * 4;
        unsafeAtomicAdd(o + 0, v * m4.x);       // -> global_atomic_add_f32
        unsafeAtomicAdd(o + 1, v * m4.y);
        unsafeAtomicAdd(o + 2, v * m4.z);
        unsafeAtomicAdd(o + 3, v * m4.w);
    }
}

// ---------------------------------------------------------------------------
// Kernel 4: leaky ReLU epilogue (bias already folded in at init)
// ---------------------------------------------------------------------------
__global__ void __launch_bounds__(256) lrelu_kernel(float* __restrict__ out, int n4) {
    int i = blockIdx.x * blockDim.x + threadIdx.x;
    if (i >= n4) return;
    float4 x = ((float4*)out)[i];
    x.x = (x.x >= 0.f) ? x.x : NEG_SLOPE * x.x;
    x.y = (x.y >= 0.f) ? x.y : NEG_SLOPE * x.y;
    x.z = (x.z >= 0.f) ? x.z : NEG_SLOPE * x.z;
    x.w = (x.w >= 0.f) ? x.w : NEG_SLOPE * x.w;
    ((float4*)out)[i] = x;
}

// ---------------------------------------------------------------------------
extern "C" void kernel_launch(void* const* d_in, const int* in_sizes, int n_in,
                              void* d_out, int out_size, void* d_ws, size_t ws_size,
                              hipStream_t stream) {
    const float* X  = (const float*)d_in[0];   // node_features [N,128]
    const int*   er = (const int*)d_in[1];     // edge_row [E]
    const int*   ec = (const int*)d_in[2];     // edge_col [E]
    const float* ev = (const float*)d_in[3];   // edge_val [E]
    const float* W  = (const float*)d_in[4];   // weight [128,128]
    const float* b  = (const float*)d_in[5];   // bias [128]
    float* out = (float*)d_out;                // [N,128]
    float* H   = (float*)d_ws;                 // scratch: h = X@W, N*128 floats

    const int n_nodes = in_sizes[0] / D_FEAT;  // 100000
    const int n_edges = in_sizes[1];           // 1600000
    const int n4      = n_nodes * (D_FEAT / 4);

    // 1) out <- broadcast(bias)
    init_bias_kernel<<<(n4 + 255) / 256, 256, 0, stream>>>(out, b, n4);
    // 2) H <- X @ W  (WMMA fp32)
    gemm_wmma_kernel<<<n_nodes / 16, 256, 0, stream>>>(X, W, H);
    // 3) out += scatter-add(edge_val * H[col])  (hardware f32 atomics)
    spmm_scatter_kernel<<<8192, 256, 0, stream>>>(H, er, ec, ev, out, n_edges);
    // 4) out <- leaky_relu(out)
    lrelu_kernel<<<(n4 + 255) / 256, 256, 0, stream>>>(out, n4);
}